// TNM_77446850282144
// MI455X (gfx1250) — compile-verified
//
#include <hip/hip_runtime.h>
#include <stdint.h>

#define B_    8
#define T_    8192
#define C_    512
#define L_    128                 // chunk length along T
#define NCH_  (T_ / L_)           // 64 chunks per sequence
#define TT_   16                  // rows per LDS sub-tile
#define NSUB_ (L_ / TT_)          // 8 sub-tiles per chunk
#define EPS_  1e-5f
#define DT_   (1.0f / 30.0f)

static_assert(L_ == 128, "dL computed as 7 squarings assumes L=128");

// ---------------- CDNA5 (gfx1250) async global->LDS helpers ----------------
#if defined(__HIP_DEVICE_COMPILE__) && __has_builtin(__builtin_amdgcn_global_load_async_to_lds_b128)
#define HAVE_ASYNC_B128 1
#else
#define HAVE_ASYNC_B128 0
#endif

typedef int v4i_ __attribute__((vector_size(16)));
typedef __attribute__((address_space(1))) v4i_ gv4i_t;   // global int4
typedef __attribute__((address_space(3))) v4i_ lv4i_t;   // LDS int4

__device__ __forceinline__ void async_ld_b128(const void* g, void* l) {
  void* gv = (void*)g;  // drop const
#if HAVE_ASYNC_B128
  __builtin_amdgcn_global_load_async_to_lds_b128(
      (gv4i_t*)gv, (lv4i_t*)l, 0, 0);
#else
  unsigned lds_off =
      (unsigned)(unsigned long long)(__attribute__((address_space(3))) char*)l;
  asm volatile("global_load_async_to_lds_b128 %0, %1, off"
               :: "v"(lds_off), "v"(gv) : "memory");
#endif
}

// waits as raw asm: guaranteed compiler memory barrier + no builtin-existence risk
#define WAIT_ASYNC(n) asm volatile("s_wait_asynccnt %0" :: "i"(n) : "memory")
#define WAIT_DS0()    asm volatile("s_wait_dscnt 0" ::: "memory")

// ---------------------------------------------------------------------------
// Chunk transfer algebra (per channel, chunk length L, decay d, om = 1-d):
//   m_out = d^L * m_in + A            A  = mu_end   (m-scan with m_in = 0)
//   diff_i = p_i - d^{i+1} * m_in     p_i = x_i - mu_i
//   v_out = d^L * v_in + P - 2*Q*m_in + R*m_in^2
//     P = sum_i om d^{L-1-i} p_i^2                      (accumulated)
//     Q = om * d^L * sum_i p_i                          (running sum only!)
//     R = d^{L+1} (1 - d^L)                             (closed form)
// Pass 1 stores (A, P, sum_p) per (b, chunk, c); Q and R are derived in scan.
// ---------------------------------------------------------------------------
__global__ void __launch_bounds__(128)
tnm_pass1(const float* __restrict__ x, const float* __restrict__ lbd,
          float4* __restrict__ apqr) {
  __shared__ float tile[2][TT_][C_];
  const int tid = threadIdx.x;        // 0..127 -> 4 channels each
  const int j   = blockIdx.x;         // chunk index
  const int b   = blockIdx.y;         // batch
  const int c4  = tid * 4;

  float lbv[4];
  *(float4*)lbv = *(const float4*)(lbd + c4);
  float dc[4], om[4], mu[4], Pa[4], Sp[4];
#pragma unroll
  for (int k = 0; k < 4; ++k) {
    dc[k] = 1.0f - 0.6931f * DT_ * __expf(-lbv[k]);
    om[k] = 1.0f - dc[k];
    mu[k] = 0.0f; Pa[k] = 0.0f; Sp[k] = 0.0f;
  }

  const float* gbase = x + ((size_t)(b * T_ + j * L_)) * C_;

  // prime buffer 0 (each lane fetches the 16B it will later read itself)
#pragma unroll
  for (int r = 0; r < TT_; ++r)
    async_ld_b128(gbase + (size_t)r * C_ + c4, &tile[0][r][c4]);

  for (int s = 0; s < NSUB_; ++s) {
    const int buf = s & 1;
    if (s + 1 < NSUB_) {
      if (s > 0) WAIT_DS0();          // our ds reads of the buffer being refilled
      const float* gn = gbase + (size_t)(s + 1) * TT_ * C_;
#pragma unroll
      for (int r = 0; r < TT_; ++r)
        async_ld_b128(gn + (size_t)r * C_ + c4, &tile[buf ^ 1][r][c4]);
      WAIT_ASYNC(TT_);                // sub-tile s complete (in-order)
    } else {
      WAIT_ASYNC(0);
    }
#pragma unroll
    for (int r = 0; r < TT_; ++r) {
      float xv[4];
      *(float4*)xv = *(const float4*)&tile[buf][r][c4];
#pragma unroll
      for (int k = 0; k < 4; ++k) {
        mu[k] = om[k] * xv[k] + dc[k] * mu[k];
        float p = xv[k] - mu[k];
        Pa[k] = om[k] * (p * p) + dc[k] * Pa[k];
        Sp[k] += p;
      }
    }
  }

  float4* o = apqr + ((size_t)(b * NCH_ + j)) * C_ + c4;
#pragma unroll
  for (int k = 0; k < 4; ++k)
    o[k] = make_float4(mu[k], Pa[k], Sp[k], 0.0f);
}

// ---------------------------------------------------------------------------
// Pass 2 (tiny): sequential scan over the 64 chunks of each (b,c) sequence.
// ---------------------------------------------------------------------------
__global__ void tnm_scan(const float* __restrict__ lbd,
                         const float* __restrict__ xmean0,
                         const float* __restrict__ x2mean0,
                         const float4* __restrict__ apqr,
                         float2* __restrict__ states) {
  int idx = blockIdx.x * blockDim.x + threadIdx.x;
  if (idx >= B_ * C_) return;
  const int b = idx / C_;
  const int c = idx - b * C_;
  const float d  = 1.0f - 0.6931f * DT_ * __expf(-lbd[c]);
  const float om = 1.0f - d;
  float dL = d;
#pragma unroll
  for (int i = 0; i < 7; ++i) dL *= dL;          // d^128 by squaring
  const float qscale = om * dL;                  // Q = qscale * sum_p
  const float R      = d * dL * (1.0f - dL);     // closed form
  float m = xmean0[c];
  float v = __expf(x2mean0[c]);
  const float4* ap = apqr   + (size_t)b * NCH_ * C_ + c;
  float2*       st = states + (size_t)b * NCH_ * C_ + c;
  for (int jj = 0; jj < NCH_; ++jj) {
    float4 f = ap[(size_t)jj * C_];              // (A, P, sum_p, -)
    st[(size_t)jj * C_] = make_float2(m, v);     // state entering chunk jj
    float Q  = qscale * f.z;
    float vn = dL * v + (f.y - 2.0f * Q * m + R * (m * m));
    m = f.x + dL * m;
    v = fmaxf(vn, 0.0f);                         // exact math is >=0; clamp fp noise
  }
}

// ---------------------------------------------------------------------------
// Pass 3: re-stream x with correct chunk-entry state; emit y = norm*gamma+beta.
// Same async double-buffered LDS pipeline as pass 1.
// ---------------------------------------------------------------------------
__global__ void __launch_bounds__(128)
tnm_pass2(const float* __restrict__ x, const float* __restrict__ lbd,
          const float* __restrict__ gamma, const float* __restrict__ beta,
          const float2* __restrict__ states, float* __restrict__ y) {
  __shared__ float tile[2][TT_][C_];
  const int tid = threadIdx.x;
  const int j   = blockIdx.x;
  const int b   = blockIdx.y;
  const int c4  = tid * 4;

  float lbv[4], gm[4], bt[4], dc[4], om[4], m[4], v[4];
  *(float4*)lbv = *(const float4*)(lbd + c4);
  *(float4*)gm  = *(const float4*)(gamma + c4);
  *(float4*)bt  = *(const float4*)(beta + c4);
  const float2* st = states + ((size_t)(b * NCH_ + j)) * C_ + c4;
#pragma unroll
  for (int k = 0; k < 4; ++k) {
    dc[k] = 1.0f - 0.6931f * DT_ * __expf(-lbv[k]);
    om[k] = 1.0f - dc[k];
    float2 s2 = st[k];
    m[k] = s2.x; v[k] = s2.y;
  }

  const float* gbase = x + ((size_t)(b * T_ + j * L_)) * C_;
  float*       ybase = y + ((size_t)(b * T_ + j * L_)) * C_;

#pragma unroll
  for (int r = 0; r < TT_; ++r)
    async_ld_b128(gbase + (size_t)r * C_ + c4, &tile[0][r][c4]);

  for (int s = 0; s < NSUB_; ++s) {
    const int buf = s & 1;
    if (s + 1 < NSUB_) {
      if (s > 0) WAIT_DS0();
      const float* gn = gbase + (size_t)(s + 1) * TT_ * C_;
#pragma unroll
      for (int r = 0; r < TT_; ++r)
        async_ld_b128(gn + (size_t)r * C_ + c4, &tile[buf ^ 1][r][c4]);
      WAIT_ASYNC(TT_);
    } else {
      WAIT_ASYNC(0);
    }
#pragma unroll
    for (int r = 0; r < TT_; ++r) {
      float xv[4], yv[4];
      *(float4*)xv = *(const float4*)&tile[buf][r][c4];
#pragma unroll
      for (int k = 0; k < 4; ++k) {
        m[k] = om[k] * xv[k] + dc[k] * m[k];     // m_new
        float d2 = xv[k] - m[k];
        v[k] = om[k] * (d2 * d2) + dc[k] * v[k]; // v_new
        yv[k] = d2 * rsqrtf(v[k] + EPS_) * gm[k] + bt[k];
      }
      *(float4*)(ybase + (size_t)(s * TT_ + r) * C_ + c4) = *(float4*)yv;
    }
  }
}

// ---------------------------------------------------------------------------
extern "C" void kernel_launch(void* const* d_in, const int* in_sizes, int n_in,
                              void* d_out, int out_size, void* d_ws, size_t ws_size,
                              hipStream_t stream) {
  (void)in_sizes; (void)n_in; (void)out_size; (void)ws_size;
  const float* x       = (const float*)d_in[0];
  const float* lbd     = (const float*)d_in[1];
  const float* gamma   = (const float*)d_in[2];
  const float* beta    = (const float*)d_in[3];
  const float* xmean0  = (const float*)d_in[4];
  const float* x2mean0 = (const float*)d_in[5];
  float* y = (float*)d_out;

  // workspace: 4 MiB of float4 (A,P,sum_p) + 2 MiB of float2 (m_in, v_in)
  float4* apqr   = (float4*)d_ws;
  float2* states = (float2*)((char*)d_ws + sizeof(float4) * (size_t)B_ * NCH_ * C_);

  dim3 grid(NCH_, B_);   // 64 x 8 = 512 blocks of 128 threads (2048 waves)
  dim3 blk(128);
  tnm_pass1<<<grid, blk, 0, stream>>>(x, lbd, apqr);
  tnm_scan<<<dim3((B_ * C_ + 255) / 256), dim3(256), 0, stream>>>(
      lbd, xmean0, x2mean0, apqr, states);
  tnm_pass2<<<grid, blk, 0, stream>>>(x, lbd, gamma, beta, states, y);
}